// GELU272_23648089932100
// MI455X (gfx1250) — compile-verified
//
#include <hip/hip_runtime.h>

typedef float v2f __attribute__((ext_vector_type(2)));
typedef float v8f __attribute__((ext_vector_type(8)));

#define GELU_C 0.7978845608028654f
#define FIRE_THRESH 0.85f

__device__ __forceinline__ float gelu_f(float v) {
    float u = GELU_C * (v + 0.044715f * v * v * v);
    return 0.5f * v * (1.0f + tanhf(u));
}

// ---------------- K1: streaming stats pass over x ----------------
// grid = (D/256, rows/256), block = 256. Each thread: one column, 256 rows.
__global__ void k1_stats(const float* __restrict__ x, float* __restrict__ colpart,
                         float* __restrict__ bsum, float* __restrict__ bsq, int D) {
    const int ROWS = 256;
    int col   = blockIdx.x * blockDim.x + threadIdx.x;
    int chunk = blockIdx.y;
    const float* p = x + (size_t)chunk * ROWS * D + col;
    float cs = 0.0f, sq = 0.0f;
    #pragma unroll 4
    for (int r = 0; r < ROWS; ++r) {
        float y = gelu_f(p[(size_t)r * D]);
        cs += y;
        sq += y * y;
    }
    colpart[(size_t)chunk * D + col] = cs;

    __shared__ float sh[256];
    int t = threadIdx.x;
    sh[t] = cs; __syncthreads();
    for (int s = 128; s > 0; s >>= 1) { if (t < s) sh[t] += sh[t + s]; __syncthreads(); }
    if (t == 0) bsum[chunk * gridDim.x + blockIdx.x] = sh[0];
    __syncthreads();
    sh[t] = sq; __syncthreads();
    for (int s = 128; s > 0; s >>= 1) { if (t < s) sh[t] += sh[t + s]; __syncthreads(); }
    if (t == 0) bsq[chunk * gridDim.x + blockIdx.x] = sh[0];
}

// ---------------- K2: deterministic final reductions ----------------
__global__ void k2_reduce(const float* __restrict__ colpart, const float* __restrict__ bsum,
                          const float* __restrict__ bsq, float* __restrict__ mean_now,
                          float* __restrict__ qv, float* __restrict__ scal,
                          int chunks, int D, int nblk, float rowsf) {
    __shared__ float  sh[256];
    __shared__ double dsh[256];
    int t = threadIdx.x;
    float inv_rows = 1.0f / rowsf;

    // per-column mean + partial ||mean||^2
    float nrm = 0.0f;
    for (int col = t; col < D; col += 256) {
        float s = 0.0f;
        for (int c = 0; c < chunks; ++c) s += colpart[(size_t)c * D + col];
        float mn = s * inv_rows;
        mean_now[col] = mn;
        nrm += mn * mn;
    }
    sh[t] = nrm; __syncthreads();
    for (int s = 128; s > 0; s >>= 1) { if (t < s) sh[t] += sh[t + s]; __syncthreads(); }
    float rn = 1.0f / fmaxf(sqrtf(sh[0]), 1e-12f);
    for (int col = t; col < D; col += 256) qv[col] = mean_now[col] * rn;

    // global sum / sumsq in double (33.5M elements)
    double ds = 0.0, dq = 0.0;
    for (int i = t; i < nblk; i += 256) { ds += (double)bsum[i]; dq += (double)bsq[i]; }
    dsh[t] = ds; __syncthreads();
    for (int s = 128; s > 0; s >>= 1) { if (t < s) dsh[t] += dsh[t + s]; __syncthreads(); }
    double tot = dsh[0]; __syncthreads();
    dsh[t] = dq; __syncthreads();
    for (int s = 128; s > 0; s >>= 1) { if (t < s) dsh[t] += dsh[t + s]; __syncthreads(); }
    if (t == 0) {
        double n   = (double)rowsf * (double)D;
        double var = (dsh[0] - tot * tot / n) / (n - 1.0);
        float stdv = sqrtf(fmaxf((float)var, 0.0f));
        scal[0] = fmaxf(stdv, 1e-5f);   // std_now = max(std, EPS)
    }
}

// ---------------- K3: WMMA GEMV  sims = buf_keys @ q ----------------
// One wave per 16 key rows. A = 16x4 f32 tile, B broadcasts q[k..k+3] to all
// 16 columns -> every output column of the 16x16 f32 accumulator equals the
// row dot product. 4 accumulators hide WMMA->WMMA RAW latency.
#define WMMA_STEP(kk, acc) {                                                   \
    float2 av = *(const float2*)(arow + (kk) + kofs);                          \
    float2 bv = *(const float2*)(q + (kk) + kofs);                             \
    v2f a, b; a.x = av.x; a.y = av.y; b.x = bv.x; b.y = bv.y;                  \
    acc = __builtin_amdgcn_wmma_f32_16x16x4_f32(false, a, false, b,            \
                                                (short)0, acc, false, false); }

__global__ void k3_sims_wmma(const float* __restrict__ keys, const float* __restrict__ q,
                             const unsigned char* __restrict__ mask,
                             float* __restrict__ sims, int D) {
    int lane = threadIdx.x & 31;
    int r0   = blockIdx.x * 16;
    int kofs = (lane & 16) ? 2 : 0;          // lanes 0-15: K 0,1 ; lanes 16-31: K 2,3
    const float* arow = keys + (size_t)(r0 + (lane & 15)) * D;

    v8f c0 = {}, c1 = {}, c2 = {}, c3 = {};
    for (int k = 0; k < D; k += 16) {
        WMMA_STEP(k,      c0);
        WMMA_STEP(k + 4,  c1);
        WMMA_STEP(k + 8,  c2);
        WMMA_STEP(k + 12, c3);
    }
    v8f c = c0 + c1 + c2 + c3;

    // D layout: lanes 0-15 VGPR j -> M=j ; lanes 16-31 VGPR j -> M=j+8.
    if (lane == 0 || lane == 16) {
        int mbase = r0 + ((lane == 16) ? 8 : 0);
        for (int j = 0; j < 8; ++j) {
            int r = mbase + j;
            sims[r] = mask[r] ? c[j] : -1.0f;
        }
    }
}

// ---------------- K4: argmax (first-index tie-break) + scalar math ----------------
__global__ void k4_finalize(const float* __restrict__ sims, const long long* __restrict__ hit_count,
                            const float* __restrict__ buf_std, const float* __restrict__ log_k_blend,
                            float* __restrict__ scal, int N) {
    __shared__ float sv[256];
    __shared__ int   si[256];
    int t = threadIdx.x;
    float best = -3.0e38f; int bi = 0x7fffffff;
    for (int i = t; i < N; i += 256) {
        float v = sims[i];
        if (v > best) { best = v; bi = i; }   // keeps lowest index on tie
    }
    sv[t] = best; si[t] = bi;
    __syncthreads();
    for (int s = 128; s > 0; s >>= 1) {
        if (t < s) {
            if (sv[t + s] > sv[t] || (sv[t + s] == sv[t] && si[t + s] < si[t])) {
                sv[t] = sv[t + s]; si[t] = si[t + s];
            }
        }
        __syncthreads();
    }
    if (t == 0) {
        int idx = si[0];
        float smax = sv[0];
        long long hit = (smax > FIRE_THRESH) ? 1 : 0;
        float count = (float)(hit_count[idx] + hit);
        float kb = expf(log_k_blend[0]);
        kb = fminf(fmaxf(kb, 0.01f), 5.0f);
        float z = kb * logf(count + 1.0f) - 0.5f;
        float blend = 1.0f / (1.0f + expf(-z));
        float std_now = scal[0];
        scal[1] = blend;
        scal[2] = buf_std[idx] / (std_now + 1e-5f);   // scale_ratio
        ((int*)scal)[3] = idx;
    }
}

// ---------------- K5: streaming output pass ----------------
// Each block: 4096 contiguous elements (= 2 full rows of D=2048); mean/mu1 in LDS.
__global__ void k5_out(const float* __restrict__ x, float* __restrict__ out,
                       const float* __restrict__ mean_now, const float* __restrict__ buf_mean,
                       const float* __restrict__ scal, int D) {
    __shared__ float smean[2048];
    __shared__ float smu[2048];
    int t = threadIdx.x;
    int idx     = ((const int*)scal)[3];
    float blend = scal[1];
    float scale = scal[2];
    const float* mu1 = buf_mean + (size_t)idx * D;
    for (int i = t; i < D; i += 256) { smean[i] = mean_now[i]; smu[i] = mu1[i]; }
    __syncthreads();

    bool  plain = (blend < 1e-6f);
    float omb   = 1.0f - blend;
    size_t base = (size_t)blockIdx.x * 4096;
    const float4* xv = (const float4*)(x + base);
    float4*       ov = (float4*)(out + base);

    for (int i = t; i < 1024; i += 256) {
        float4 v = xv[i];
        int col = (i * 4) % D;
        float y0 = gelu_f(v.x), y1 = gelu_f(v.y), y2 = gelu_f(v.z), y3 = gelu_f(v.w);
        float4 r;
        if (plain) {
            r.x = y0; r.y = y1; r.z = y2; r.w = y3;
        } else {
            r.x = omb * y0 + blend * (smu[col + 0] + (y0 - smean[col + 0]) * scale);
            r.y = omb * y1 + blend * (smu[col + 1] + (y1 - smean[col + 1]) * scale);
            r.z = omb * y2 + blend * (smu[col + 2] + (y2 - smean[col + 2]) * scale);
            r.w = omb * y3 + blend * (smu[col + 3] + (y3 - smean[col + 3]) * scale);
        }
        ov[i] = r;
    }
}

extern "C" void kernel_launch(void* const* d_in, const int* in_sizes, int n_in,
                              void* d_out, int out_size, void* d_ws, size_t ws_size,
                              hipStream_t stream) {
    const float*         x           = (const float*)d_in[0];
    const float*         log_k_blend = (const float*)d_in[1];
    const float*         buf_keys    = (const float*)d_in[2];
    const float*         buf_mean    = (const float*)d_in[3];
    const float*         buf_std     = (const float*)d_in[4];
    const unsigned char* mask        = (const unsigned char*)d_in[5]; // numpy bool
    const long long*     hit_count   = (const long long*)d_in[6];     // int64
    float* out = (float*)d_out;

    int N    = in_sizes[4];                 // 512
    int D    = in_sizes[2] / N;             // 2048
    int rows = in_sizes[0] / D;             // 16384 (= B*T)
    int chunks = rows / 256;                // 64
    int cgroups = D / 256;                  // 8
    int nblk = chunks * cgroups;            // 512

    // workspace layout (floats)
    float* ws       = (float*)d_ws;
    float* colpart  = ws;                                  // chunks * D
    float* bsum     = colpart + (size_t)chunks * D;        // nblk
    float* bsq      = bsum + nblk;                         // nblk
    float* mean_now = bsq + nblk;                          // D
    float* qv       = mean_now + D;                        // D
    float* sims     = qv + D;                              // N
    float* scal     = sims + N;                            // 8 scalars

    dim3 g1(cgroups, chunks);
    k1_stats<<<g1, 256, 0, stream>>>(x, colpart, bsum, bsq, D);
    k2_reduce<<<1, 256, 0, stream>>>(colpart, bsum, bsq, mean_now, qv, scal,
                                     chunks, D, nblk, (float)rows);
    k3_sims_wmma<<<N / 16, 32, 0, stream>>>(buf_keys, qv, mask, sims, D);
    k4_finalize<<<1, 256, 0, stream>>>(sims, hit_count, buf_std, log_k_blend, scal, N);

    size_t total = (size_t)rows * (size_t)D;
    k5_out<<<(unsigned)(total / 4096), 256, 0, stream>>>(x, out, mean_now, buf_mean, scal, D);
}